// SNNL_20512763806274
// MI455X (gfx1250) — compile-verified
//
#include <hip/hip_runtime.h>
#include <hip/hip_bf16.h>

typedef __attribute__((ext_vector_type(2))) float v2f;
typedef __attribute__((ext_vector_type(8))) float v8f;

#define N_PTS 4608          // B*h*w = 2*48*48
#define CF 256              // feature channels
#define HH 384
#define WW 384
#define hh 48
#define ww 48
#define TILE 16
#define ROWS_PER_BLOCK 128  // 8 waves * 16 rows
#define NI_BLOCKS (N_PTS / ROWS_PER_BLOCK)   // 36
#define NJ_TILES  (N_PTS / TILE)             // 288
#define JSPLIT 8
#define JT_PER_BLOCK (NJ_TILES / JSPLIT)     // 36
#define LDS_STRIDE 260      // floats; 4-bank row skew -> conflict free

// ---------------------------------------------------------------------------
// Kernel 1: per-row squared norms, downsampled labels, zero accumulators.
// One wave32 per row.
// ---------------------------------------------------------------------------
__global__ void __launch_bounds__(256) snnl_prep(
    const int* __restrict__ labels, const float* __restrict__ X,
    float* __restrict__ sq, int* __restrict__ y,
    float* __restrict__ num, float* __restrict__ den) {
  const int row  = (int)(blockIdx.x * blockDim.x + threadIdx.x) >> 5;
  const int lane = (int)threadIdx.x & 31;
  if (row >= N_PTS) return;

  const float* xr = X + (size_t)row * CF;
  float s = 0.0f;
#pragma unroll
  for (int k = lane; k < CF; k += 32) { float v = xr[k]; s += v * v; }
#pragma unroll
  for (int m = 16; m > 0; m >>= 1) s += __shfl_xor(s, m, 32);

  if (lane == 0) {
    sq[row] = s;
    const int b   = row / (hh * ww);
    const int rem = row % (hh * ww);
    const int r   = rem / ww;
    const int c   = rem % ww;
    const int iy  = (r * HH) / hh;   // == 8*r (nearest, floor)
    const int ix  = (c * WW) / ww;   // == 8*c
    y[row]  = labels[((size_t)b * HH + iy) * WW + ix];
    num[row] = 0.0f;
    den[row] = 0.0f;
  }
}

// ---------------------------------------------------------------------------
// Kernel 2: fused  G = X X^T  (f32 WMMA 16x16x4) -> exp(-d) -> masked row
// sums.  Each wave owns one 16-row A-panel (64 v2f fragments in VGPRs) and
// streams 16-row J-tiles through a double-buffered LDS pipeline: next tile's
// global loads are issued before the current tile's 64 WMMAs, so L2 latency
// hides under the matrix pipe; only LDS stores + one barrier stay exposed.
// ---------------------------------------------------------------------------
__global__ void __launch_bounds__(256) snnl_main(
    const float* __restrict__ X, const float* __restrict__ sq,
    const int* __restrict__ y,
    float* __restrict__ num, float* __restrict__ den) {
  __shared__ float ldsJ[2][TILE * LDS_STRIDE];

  const int tid  = (int)threadIdx.x;
  const int ib   = (int)blockIdx.x / JSPLIT;
  const int js   = (int)blockIdx.x % JSPLIT;
  const int wv   = tid >> 5;
  const int lane = tid & 31;
  const int hi   = lane >> 4;      // half-wave select (K pair / M+8)
  const int l15  = lane & 15;

  const int I0 = ib * ROWS_PER_BLOCK + wv * TILE;

  // Staging geometry (per thread: 4 float4 = 64B of the 16KB tile).
  int srow[4], scol[4];
#pragma unroll
  for (int it = 0; it < 4; ++it) {
    const int idx = tid + it * 256;
    srow[it] = idx >> 6;   // 64 float4 per row
    scol[it] = idx & 63;
  }

  // --- A fragments: X[I0+l15][4kk + 2*hi .. +1], kept in registers --------
  v2f afrag[CF / 4];
  {
    const float* arow = X + (size_t)(I0 + l15) * CF + 2 * hi;
#pragma unroll
    for (int kk = 0; kk < CF / 4; ++kk)
      afrag[kk] = *(const v2f*)(arow + 4 * kk);
  }

  // Per-lane row metadata for the 8 output rows this lane touches.
  float sq_i[8]; int y_i[8];
#pragma unroll
  for (int v = 0; v < 8; ++v) {
    const int i = I0 + v + 8 * hi;
    sq_i[v] = sq[i];
    y_i[v]  = y[i];
  }

  float racc_num[8], racc_den[8];
#pragma unroll
  for (int v = 0; v < 8; ++v) { racc_num[v] = 0.0f; racc_den[v] = 0.0f; }

  // --- Prologue: stage tile 0 into buffer 0 -------------------------------
  {
    const int J0 = js * JT_PER_BLOCK * TILE;
    const float4* src = (const float4*)(X + (size_t)J0 * CF);
#pragma unroll
    for (int it = 0; it < 4; ++it)
      *(float4*)(&ldsJ[0][srow[it] * LDS_STRIDE + scol[it] * 4]) =
          src[srow[it] * 64 + scol[it]];
  }
  __syncthreads();

  int buf = 0;
  for (int jt = 0; jt < JT_PER_BLOCK; ++jt) {
    const int J0 = (js * JT_PER_BLOCK + jt) * TILE;

    // Issue next tile's global loads now; they complete under the WMMAs.
    float4 stg[4];
    if (jt + 1 < JT_PER_BLOCK) {
      const float4* src = (const float4*)(X + (size_t)(J0 + TILE) * CF);
#pragma unroll
      for (int it = 0; it < 4; ++it) stg[it] = src[srow[it] * 64 + scol[it]];
    }

    // --- 16x16 Gram tile: 64 x v_wmma_f32_16x16x4_f32 --------------------
    v8f acc = {};
    const float* brow = &ldsJ[buf][l15 * LDS_STRIDE + 2 * hi];
#pragma unroll
    for (int kk = 0; kk < CF / 4; ++kk) {
      v2f b = *(const v2f*)(brow + 4 * kk);
      acc = __builtin_amdgcn_wmma_f32_16x16x4_f32(
          false, afrag[kk], false, b, (short)0, acc, false, false);
    }

    // --- Fused epilogue: d -> exp(-d) -> masks -> row partial sums -------
    const int   j   = J0 + l15;          // column this lane holds
    const float sqj = sq[j];
    const int   yj  = y[j];
#pragma unroll
    for (int v = 0; v < 8; ++v) {
      const int i = I0 + v + 8 * hi;
      float d = sq_i[v] + sqj - 2.0f * acc[v];
      d = fmaxf(d, 0.0f);
      float e = __expf(-d);
      e = (i == j) ? 0.0f : e;           // exclude diagonal
      racc_den[v] += e;
      racc_num[v] += (y_i[v] == yj) ? e : 0.0f;
    }

    // Drain staged registers into the other LDS buffer, then one barrier.
    if (jt + 1 < JT_PER_BLOCK) {
#pragma unroll
      for (int it = 0; it < 4; ++it)
        *(float4*)(&ldsJ[buf ^ 1][srow[it] * LDS_STRIDE + scol[it] * 4]) =
            stg[it];
    }
    __syncthreads();
    buf ^= 1;
  }

  // Reduce over the 16 lanes of each half-wave (columns j mod 16) and
  // accumulate into global per-row sums.
#pragma unroll
  for (int v = 0; v < 8; ++v) {
    float n_ = racc_num[v], d_ = racc_den[v];
#pragma unroll
    for (int m = 8; m > 0; m >>= 1) {
      n_ += __shfl_xor(n_, m, 16);
      d_ += __shfl_xor(d_, m, 16);
    }
    if (l15 == 0) {
      const int i = I0 + v + 8 * hi;
      atomicAdd(&num[i], n_);
      atomicAdd(&den[i], d_);
    }
  }
}

// ---------------------------------------------------------------------------
// Kernel 3: loss = mean(log(den) - log(num))
// ---------------------------------------------------------------------------
__global__ void __launch_bounds__(256) snnl_final(
    const float* __restrict__ num, const float* __restrict__ den,
    float* __restrict__ out) {
  const int i = (int)(blockIdx.x * blockDim.x + threadIdx.x);
  float c = 0.0f;
  if (i < N_PTS) c = (__logf(den[i]) - __logf(num[i])) * (1.0f / N_PTS);
#pragma unroll
  for (int m = 16; m > 0; m >>= 1) c += __shfl_xor(c, m, 32);
  if (((int)threadIdx.x & 31) == 0) atomicAdd(out, c);
}

// ---------------------------------------------------------------------------
extern "C" void kernel_launch(void* const* d_in, const int* in_sizes, int n_in,
                              void* d_out, int out_size, void* d_ws,
                              size_t ws_size, hipStream_t stream) {
  const int*   labels = (const int*)d_in[0];   // [B,H,W]
  const float* X      = (const float*)d_in[2]; // [B,C,h,w] -> row-major [N,256]

  float* sq  = (float*)d_ws;
  int*   y   = (int*)(sq + N_PTS);
  float* num = (float*)(y + N_PTS);
  float* den = num + N_PTS;

  (void)hipMemsetAsync(d_out, 0, sizeof(float), stream);

  snnl_prep<<<N_PTS / 8, 256, 0, stream>>>(labels, X, sq, y, num, den);
  snnl_main<<<NI_BLOCKS * JSPLIT, 256, 0, stream>>>(X, sq, y, num, den);
  snnl_final<<<(N_PTS + 255) / 256, 256, 0, stream>>>(num, den, (float*)d_out);
}